// AttnDecoderGRU_82205674046075
// MI455X (gfx1250) — compile-verified
//
#include <hip/hip_runtime.h>
#include <hip/hip_bf16.h>

// ---------------------------------------------------------------------------
// AttnDecoderGRU step for MI455X (gfx1250), fp32 via V_WMMA_F32_16X16X4_F32.
// B=4096, S=200, H=512, OUT=64.
// ---------------------------------------------------------------------------

typedef __attribute__((ext_vector_type(2))) float v2f;
typedef __attribute__((ext_vector_type(8))) float v8f;

#define NEG_FLT_MAX (-3.402823466e+38f)

// ---------------------------------------------------------------------------
// cat[b, 0:OUT] = last_input[b,:] ; cat[b, OUT:OUT+H] = h0[b,:]
// ---------------------------------------------------------------------------
__global__ void build_cat_kernel(const float* __restrict__ last,
                                 const float* __restrict__ h0,
                                 float* __restrict__ cat,
                                 int B, int OUT, int H) {
    int i = blockIdx.x * blockDim.x + threadIdx.x;
    int W = OUT + H;
    if (i >= B * W) return;
    int b = i / W;
    int j = i - b * W;
    cat[i] = (j < OUT) ? last[b * OUT + j] : h0[(size_t)b * H + (j - OUT)];
}

// ---------------------------------------------------------------------------
// C[m,n] = act( sum_k A[m,k] * W[n,k] + bias[n] )   (i.e. C = A @ W^T + b)
// A: [M,K] row-major, W: [N,K] row-major, C: [M,N] row-major.
// Each wave computes a 16x64 strip (4 accumulators): one A fragment per
// K-step feeds 4 chained WMMAs -> 4x A reuse, high WMMA issue density.
//
// No masking in the K-loop: WMMA is column-separable, so out-of-range
// columns just compute garbage in lanes whose stores are skipped. Pointers
// for those lanes are clamped to W row 0 (in-bounds, finite). This keeps the
// inner loop branch-free: pure global_load_b64 + v_wmma (EXEC stays all-1s,
// as the ISA requires for WMMA).
// ---------------------------------------------------------------------------
__global__ void gemm_wmma_f32_kernel(const float* __restrict__ A,
                                     const float* __restrict__ W,
                                     const float* __restrict__ bias,
                                     float* __restrict__ C,
                                     int M, int N, int K, int relu) {
    const int wave = threadIdx.x >> 5;
    const int lane = threadIdx.x & 31;
    const int half = lane >> 4;   // 0: lanes 0-15, 1: lanes 16-31
    const int l16  = lane & 15;

    const int ntiles  = (N + 15) >> 4;
    const int nstrips = (ntiles + 3) >> 2;      // 4 n-tiles per wave
    const int mtiles  = M >> 4;
    const int strip = blockIdx.x * (blockDim.x >> 5) + wave;
    if (strip >= mtiles * nstrips) return;      // wave-uniform: no EXEC divergence

    const int tm = strip / nstrips;
    const int ts = strip - tm * nstrips;

    const int row  = tm * 16 + l16;             // A row for this lane
    const int col0 = ts * 64 + l16;             // lane's column in n-tile 0 of strip

    const float* __restrict__ arow = A + (size_t)row * K;

    bool ok[4];
    const float* __restrict__ wr[4];
#pragma unroll
    for (int j = 0; j < 4; ++j) {
        int c = col0 + 16 * j;
        ok[j] = (c < N);
        wr[j] = W + (size_t)(ok[j] ? c : 0) * K;   // clamp keeps loads in-bounds
    }

    v8f acc[4] = {};
    const int kb = half * 2;   // K sub-offset per lane half (0 or 2)

    for (int k0 = 0; k0 < K; k0 += 4) {
        // 8-byte aligned: K even, k0 multiple of 4, kb in {0,2}
        const v2f a = *(const v2f*)(arow + k0 + kb);
#pragma unroll
        for (int j = 0; j < 4; ++j) {
            const v2f b = *(const v2f*)(wr[j] + k0 + kb);
            // (neg_a, A, neg_b, B, c_mod, C, reuse_a, reuse_b)
            acc[j] = __builtin_amdgcn_wmma_f32_16x16x4_f32(
                false, a, false, b, (short)0, acc[j], false, false);
        }
    }

#pragma unroll
    for (int j = 0; j < 4; ++j) {
        if (!ok[j]) continue;
        const int col = col0 + 16 * j;
        const float bv = bias[col];
#pragma unroll
        for (int r = 0; r < 8; ++r) {
            int m = tm * 16 + r + half * 8;  // D layout: VGPR r -> row r (+8 hi lanes)
            float v = acc[j][r] + bv;
            if (relu) v = fmaxf(v, 0.0f);
            C[(size_t)m * N + col] = v;
        }
    }
}

// ---------------------------------------------------------------------------
// In-place row softmax over S elements (S <= 256). One block per row.
// ---------------------------------------------------------------------------
__global__ void softmax_rows_kernel(float* __restrict__ data, int S) {
    __shared__ float red[256];
    float* row = data + (size_t)blockIdx.x * S;
    const int t = threadIdx.x;

    float v = (t < S) ? row[t] : NEG_FLT_MAX;
    red[t] = v;
    __syncthreads();
    for (int off = 128; off > 0; off >>= 1) {
        if (t < off) red[t] = fmaxf(red[t], red[t + off]);
        __syncthreads();
    }
    const float mx = red[0];
    __syncthreads();

    float e = (t < S) ? expf(v - mx) : 0.0f;
    red[t] = e;
    __syncthreads();
    for (int off = 128; off > 0; off >>= 1) {
        if (t < off) red[t] += red[t + off];
        __syncthreads();
    }
    const float inv = 1.0f / red[0];
    if (t < S) row[t] = e * inv;
}

// ---------------------------------------------------------------------------
// attn_applied[b,h] = sum_s attw[b,s] * enc[s,b,h]; written into cat[b, OUT+h].
// enc is [S,B,H]. One block (128 threads) per b; each thread owns 4 h's.
// Streams the 1.68 GB encoder tensor -> the HBM roofline driver (~72 us floor
// at 23.3 TB/s). float4 coalesced loads + global_prefetch a few planes ahead.
// ---------------------------------------------------------------------------
__global__ void attn_apply_kernel(const float* __restrict__ attw,
                                  const float* __restrict__ enc,
                                  float* __restrict__ cat,
                                  int B, int S, int H, int CATW, int OUT) {
    __shared__ float w[256];
    const int b = blockIdx.x;
    const int t = threadIdx.x;   // 128 threads, H = 512 = 128*4

    for (int s = t; s < S; s += blockDim.x) w[s] = attw[(size_t)b * S + s];
    __syncthreads();

    const float4* __restrict__ encb =
        (const float4*)(enc + (size_t)b * H) + t;
    const size_t strideS = ((size_t)B * H) >> 2;   // float4 stride between s-planes

    float4 acc = {0.0f, 0.0f, 0.0f, 0.0f};
    for (int s = 0; s < S; ++s) {
        if (s + 6 < S)   // run the HBM stream ahead of the FMA chain
            __builtin_prefetch((const void*)(encb + (size_t)(s + 6) * strideS), 0, 0);
        const float ws = w[s];
        const float4 e = encb[(size_t)s * strideS];
        acc.x = fmaf(ws, e.x, acc.x);
        acc.y = fmaf(ws, e.y, acc.y);
        acc.z = fmaf(ws, e.z, acc.z);
        acc.w = fmaf(ws, e.w, acc.w);
    }
    float* dst = cat + (size_t)b * CATW + OUT + t * 4;
    dst[0] = acc.x; dst[1] = acc.y; dst[2] = acc.z; dst[3] = acc.w;
}

// ---------------------------------------------------------------------------
// GRU gates (PyTorch order r,z,n):
//   r = sig(gi_r+gh_r); z = sig(gi_z+gh_z); n = tanh(gi_n + r*gh_n)
//   h_new = (1-z)*n + z*h0
// ---------------------------------------------------------------------------
__global__ void gru_gate_kernel(const float* __restrict__ gi,
                                const float* __restrict__ gh,
                                const float* __restrict__ h0,
                                float* __restrict__ hnew,
                                int B, int H) {
    int i = blockIdx.x * blockDim.x + threadIdx.x;
    if (i >= B * H) return;
    int b = i / H;
    int h = i - b * H;
    size_t base = (size_t)b * 3 * H;
    float ir = gi[base + h], iz = gi[base + H + h], in = gi[base + 2 * H + h];
    float hr = gh[base + h], hz = gh[base + H + h], hn = gh[base + 2 * H + h];
    float r = 1.0f / (1.0f + expf(-(ir + hr)));
    float z = 1.0f / (1.0f + expf(-(iz + hz)));
    float n = tanhf(in + r * hn);
    hnew[i] = (1.0f - z) * n + z * h0[i];
}

// ---------------------------------------------------------------------------
// host-side launch helpers
// ---------------------------------------------------------------------------
static inline void launch_gemm(const float* A, const float* W, const float* bias,
                               float* C, int M, int N, int K, int relu,
                               hipStream_t stream) {
    const int WAVES = 4;                       // 128-thread blocks = 4 wave32
    int ntiles  = (N + 15) >> 4;
    int nstrips = (ntiles + 3) >> 2;
    int strips  = (M >> 4) * nstrips;
    int blocks  = (strips + WAVES - 1) / WAVES;
    gemm_wmma_f32_kernel<<<blocks, WAVES * 32, 0, stream>>>(A, W, bias, C, M, N, K, relu);
}

extern "C" void kernel_launch(void* const* d_in, const int* in_sizes, int n_in,
                              void* d_out, int out_size, void* d_ws, size_t ws_size,
                              hipStream_t stream) {
    (void)in_sizes; (void)n_in; (void)out_size; (void)ws_size;

    const float* last   = (const float*)d_in[0];   // [B,OUT]
    const float* hidden = (const float*)d_in[1];   // [1,B,H] -> h0
    const float* enc    = (const float*)d_in[2];   // [S,B,H]
    const float* attn_W = (const float*)d_in[3];   // [S, H+OUT]
    const float* attn_b = (const float*)d_in[4];   // [S]
    const float* comb_W = (const float*)d_in[5];   // [H, H+OUT]
    const float* comb_b = (const float*)d_in[6];   // [H]
    const float* w_ih   = (const float*)d_in[7];   // [3H, H]
    const float* w_hh   = (const float*)d_in[8];   // [3H, H]
    const float* b_ih   = (const float*)d_in[9];   // [3H]
    const float* b_hh   = (const float*)d_in[10];  // [3H]
    const float* out_W  = (const float*)d_in[11];  // [OUT, H]
    const float* out_b  = (const float*)d_in[12];  // [OUT]

    const int B = 4096, S = 200, H = 512, OUT = 64;
    const int CATW = H + OUT;                      // 576

    // Output layout (tuple flattened): out[B,OUT] | h_new[1,B,H] | attn_w[B,S]
    float* out_p  = (float*)d_out;
    float* hnew_p = out_p + (size_t)B * OUT;
    float* attw_p = hnew_p + (size_t)B * H;

    // Workspace layout
    float* cat = (float*)d_ws;                       // [B, CATW]  (cat1, then cat2)
    float* x   = cat + (size_t)B * CATW;             // [B, H]
    float* gi  = x   + (size_t)B * H;                // [B, 3H]
    float* gh  = gi  + (size_t)B * 3 * H;            // [B, 3H]

    // 1) cat1 = [last_input | h0]
    build_cat_kernel<<<(B * CATW + 255) / 256, 256, 0, stream>>>(
        last, hidden, cat, B, OUT, H);

    // 2) attention logits -> attn_weights output region (pre-softmax)
    launch_gemm(cat, attn_W, attn_b, attw_p, B, S, CATW, /*relu=*/0, stream);

    // 3) softmax over S, in place
    softmax_rows_kernel<<<B, 256, 0, stream>>>(attw_p, S);

    // 4) attn_applied -> cat[:, OUT:]  (cat becomes cat2; last_input prefix shared)
    attn_apply_kernel<<<B, 128, 0, stream>>>(attw_p, enc, cat, B, S, H, CATW, OUT);

    // 5) x = relu(cat2 @ comb_W^T + comb_b)
    launch_gemm(cat, comb_W, comb_b, x, B, H, CATW, /*relu=*/1, stream);

    // 6) gi = x @ w_ih^T + b_ih
    launch_gemm(x, w_ih, b_ih, gi, B, 3 * H, H, /*relu=*/0, stream);

    // 7) gh = h0 @ w_hh^T + b_hh
    launch_gemm(hidden, w_hh, b_hh, gh, B, 3 * H, H, /*relu=*/0, stream);

    // 8) GRU gate fusion -> h_new output region
    gru_gate_kernel<<<(B * H + 255) / 256, 256, 0, stream>>>(
        gi, gh, hidden, hnew_p, B, H);

    // 9) out = h_new @ out_W^T + out_b
    launch_gemm(hnew_p, out_W, out_b, out_p, B, OUT, H, /*relu=*/0, stream);
}